// FC_1236950581476
// MI455X (gfx1250) — compile-verified
//
#include <hip/hip_runtime.h>

typedef __attribute__((ext_vector_type(2))) float v2f;
typedef __attribute__((ext_vector_type(4))) float v4f;
typedef __attribute__((ext_vector_type(8))) float v8f;
typedef __attribute__((ext_vector_type(4))) int   v4i;

// global (addrspace 1) pointer so table gathers emit global_load_b32, not flat_load
typedef const float __attribute__((address_space(1)))* gfptr;

#define NROWS 524288
#define NPTN  48
#define NVAL  16
#define XS    68        // padded LDS row stride in floats (even, 272B: rows 16B-aligned)
#define WPB   8         // waves per block
#define TPW   4         // tiles per wave
#define NBLK  (NROWS / 16 / WPB / TPW)   // 1024
#define NWAVE (NBLK * WPB)               // 8192 waves

struct Tabs { const float* t[13]; };
// t[] index: 0=e1 1=c52 2=c33 3=e2 4=e3 5=e4 6=k8 7=k7 8=k6 9=k5 10=k4 11=ccor 12=cx22

__constant__ unsigned char col2tab[48] = {
    3,3,3,3,  4,4,4,4,  5,5,5,5,  6,6,  7,7,7,7,  8,8,8,8,
    9,9,9,9,  10,10,10,10, 11, 12,
    0, 2, 1, 2, 1, 0, 1, 0, 1, 0, 2, 1, 2, 1, 1, 1
};

__device__ __forceinline__ const float* tab_ptr(const Tabs& tb, int id) {
  switch (id) {
    case 0:  return tb.t[0];
    case 1:  return tb.t[1];
    case 2:  return tb.t[2];
    case 3:  return tb.t[3];
    case 4:  return tb.t[4];
    case 5:  return tb.t[5];
    case 6:  return tb.t[6];
    case 7:  return tb.t[7];
    case 8:  return tb.t[8];
    case 9:  return tb.t[9];
    case 10: return tb.t[10];
    case 11: return tb.t[11];
    default: return tb.t[12];
  }
}

__device__ __forceinline__ float relu(float x) {
  return __builtin_amdgcn_fmed3f(x, 0.0f, __builtin_inff());
}

__global__ __launch_bounds__(256) void mlp_gather_wmma(
    const float* __restrict__ xvalue,
    const int*   __restrict__ xboard,
    Tabs tabs,
    const float* __restrict__ W1, const float* __restrict__ b1,
    const float* __restrict__ W2, const float* __restrict__ b2,
    const float* __restrict__ Wo, const float* __restrict__ bo,
    float* __restrict__ out)
{
  __shared__ float              xt[WPB][16 * XS];  // per-wave 16x64 activation tile
  __shared__ unsigned long long tptr[NPTN];        // raw table base addresses

  const int tid  = threadIdx.x;
  const int lane = tid & 31;
  const int wv   = tid >> 5;
  const int half = lane >> 4;     // 0: lanes 0-15, 1: lanes 16-31
  const int n    = lane & 15;     // column (B/C/D) & row (A) index for this lane

  if (tid < NPTN)
    tptr[tid] = (unsigned long long)tab_ptr(tabs, (int)col2tab[tid]);
  __syncthreads();

  float* xw = xt[wv];
  const long wave_id = (long)blockIdx.x * WPB + wv;

  // ---- hoisted weight fragments (reused across TPW tiles) ----
  // B-matrix layout: lane n holds W[k0..k0+1][n], k0 = 4c + 2*half
  v2f w1f[16];
  #pragma unroll
  for (int c = 0; c < 16; ++c) {
    const int k0 = 4 * c + 2 * half;
    w1f[c].x = W1[k0 * 16 + n];
    w1f[c].y = W1[(k0 + 1) * 16 + n];
  }
  v2f w2f[4];
  #pragma unroll
  for (int c = 0; c < 4; ++c) {
    const int k0 = 4 * c + 2 * half;
    w2f[c].x = W2[k0 * 16 + n];
    w2f[c].y = W2[(k0 + 1) * 16 + n];
  }
  // Wo as 16x16 B-matrix with only column 0 nonzero: fragment is zero for n != 0
  v2f wof[4];
  #pragma unroll
  for (int c = 0; c < 4; ++c) {
    const int k0 = 4 * c + 2 * half;
    wof[c].x = (n == 0) ? Wo[k0]     : 0.0f;
    wof[c].y = (n == 0) ? Wo[k0 + 1] : 0.0f;
  }
  // loop-invariant bias splats, consumed directly as C by the first WMMA of each chain
  v8f c1, c2, c3;
  {
    const float b1n = b1[n];
    const float b2n = b2[n];
    const float bon = bo[0];
    #pragma unroll
    for (int r = 0; r < 8; ++r) { c1[r] = b1n; c2[r] = b2n; c3[r] = bon; }
  }

  for (int it = 0; it < TPW; ++it) {
    const long tile = wave_id + (long)it * NWAVE;   // grid-stride: contiguous sweep/iter
    const long r0   = tile * 16;

    // ---- stage xvalue (16 rows x 16 f32, contiguous 1KB) into LDS cols [0,16) ----
    {
      const v4f* src = (const v4f*)(xvalue + r0 * NVAL);
      #pragma unroll
      for (int j = 0; j < 2; ++j) {
        const int q   = lane * 2 + j;                        // float4 chunk 0..63
        const v4f v   = __builtin_nontemporal_load(src + q); // TH=NT: keep L2 for tables
        const int f   = q * 4;                               // flat element 0..255
        const int row = f >> 4;
        const int col = f & 15;
        *(v4f*)&xw[row * XS + col] = v;
      }
    }

    // ---- gather 16x48 table values into LDS cols [16,64) ----
    {
      const v4i* bsrc = (const v4i*)(xboard + r0 * NPTN);    // 768 contiguous ints
      #pragma unroll
      for (int j = 0; j < 6; ++j) {
        const int q    = j * 32 + lane;                      // int4 chunk 0..191
        const v4i idx4 = __builtin_nontemporal_load(bsrc + q);
        const int f0   = q * 4;
        #pragma unroll
        for (int e = 0; e < 4; ++e) {
          const int f   = f0 + e;
          const int row = f / NPTN;
          const int col = f - row * NPTN;
          const gfptr base = (gfptr)tptr[col];               // global_load_b32 gather
          xw[row * XS + 16 + col] = base[idx4[e]];
        }
      }
    }
    __builtin_amdgcn_wave_barrier();  // same-wave LDS ops are in-order; fence compiler

    // ---- layer 1: 16x64 @ 64x16 via 16 x V_WMMA_F32_16X16X4_F32, C seeded with b1 ----
    v8f acc;
    {
      const v2f a = *(const v2f*)&xw[n * XS + 2 * half];
      acc = __builtin_amdgcn_wmma_f32_16x16x4_f32(
          false, a, false, w1f[0], (short)0, c1, false, false);
    }
    #pragma unroll
    for (int c = 1; c < 16; ++c) {
      const int c0 = 4 * c + 2 * half;                       // even -> 8B-aligned LDS b64
      const v2f a = *(const v2f*)&xw[n * XS + c0];
      acc = __builtin_amdgcn_wmma_f32_16x16x4_f32(
          false, a, false, w1f[c], (short)0, acc, false, false);
    }
    #pragma unroll
    for (int r = 0; r < 8; ++r) acc[r] = relu(acc[r]);

    // ---- reshape h (C/D layout) -> A layout through this wave's LDS region ----
    #pragma unroll
    for (int r = 0; r < 8; ++r) xw[(r + 8 * half) * XS + n] = acc[r];
    __builtin_amdgcn_wave_barrier();

    // ---- layer 2: 16x16 @ 16x16 via 4 WMMAs, C seeded with b2 ----
    v8f acc2;
    {
      const v2f a = *(const v2f*)&xw[n * XS + 2 * half];
      acc2 = __builtin_amdgcn_wmma_f32_16x16x4_f32(
          false, a, false, w2f[0], (short)0, c2, false, false);
    }
    #pragma unroll
    for (int c = 1; c < 4; ++c) {
      const int c0 = 4 * c + 2 * half;
      const v2f a = *(const v2f*)&xw[n * XS + c0];
      acc2 = __builtin_amdgcn_wmma_f32_16x16x4_f32(
          false, a, false, w2f[c], (short)0, acc2, false, false);
    }
    #pragma unroll
    for (int r = 0; r < 8; ++r) acc2[r] = relu(acc2[r]);

    // ---- reshape relu(h2) -> A layout (same LDS slots) ----
    #pragma unroll
    for (int r = 0; r < 8; ++r) xw[(r + 8 * half) * XS + n] = acc2[r];
    __builtin_amdgcn_wave_barrier();

    // ---- output layer as WMMA: D = relu(h2) @ WoMat + bo; column 0 of D = result ----
    v8f acc3;
    {
      const v2f a = *(const v2f*)&xw[n * XS + 2 * half];
      acc3 = __builtin_amdgcn_wmma_f32_16x16x4_f32(
          false, a, false, wof[0], (short)0, c3, false, false);
    }
    #pragma unroll
    for (int c = 1; c < 4; ++c) {
      const int c0 = 4 * c + 2 * half;
      const v2f a = *(const v2f*)&xw[n * XS + c0];
      acc3 = __builtin_amdgcn_wmma_f32_16x16x4_f32(
          false, a, false, wof[c], (short)0, acc3, false, false);
    }

    // D layout: lane 0 VGPR r = out[row r], lane 16 VGPR r = out[row r+8]
    if (n == 0) {
      v4f o0 = {acc3[0], acc3[1], acc3[2], acc3[3]};
      v4f o1 = {acc3[4], acc3[5], acc3[6], acc3[7]};
      v4f* dst = (v4f*)(out + r0 + 8 * half);
      __builtin_nontemporal_store(o0, dst);     // streamed once: TH=NT
      __builtin_nontemporal_store(o1, dst + 1);
    }
    __builtin_amdgcn_wave_barrier();  // keep next iter's LDS stores ordered after reads
  }
}

extern "C" void kernel_launch(void* const* d_in, const int* in_sizes, int n_in,
                              void* d_out, int out_size, void* d_ws, size_t ws_size,
                              hipStream_t stream) {
  (void)in_sizes; (void)n_in; (void)out_size; (void)d_ws; (void)ws_size;

  const float* xvalue = (const float*)d_in[0];
  const int*   xboard = (const int*)d_in[1];

  Tabs tabs;
  for (int i = 0; i < 13; ++i) tabs.t[i] = (const float*)d_in[2 + i];

  const float* W1 = (const float*)d_in[15];
  const float* b1 = (const float*)d_in[16];
  const float* W2 = (const float*)d_in[17];
  const float* b2 = (const float*)d_in[18];
  const float* Wo = (const float*)d_in[19];
  const float* bo = (const float*)d_in[20];
  float* out = (float*)d_out;

  mlp_gather_wmma<<<NBLK, WPB * 32, 0, stream>>>(
      xvalue, xboard, tabs, W1, b1, W2, b2, Wo, bo, out);
}